// MultiHeadAttension_1503238554059
// MI455X (gfx1250) — compile-verified
//
#include <hip/hip_runtime.h>
#include <math.h>

typedef __attribute__((ext_vector_type(2))) float v2f;
typedef __attribute__((ext_vector_type(4))) float v4f;
typedef __attribute__((ext_vector_type(8))) float v8f;

#define TSEQ 2048
#define CCH  192
#define NH   2
#define KH   96
#define BAND 256
#define WCOL 528      // 2*BAND + 16 (aligned band window per 16-row tile)
#define NJT  33       // WCOL / 16
#define QSCALE 0.10206207261596575f  // 1/sqrt(96)

__device__ __forceinline__ v8f wmma_f32x4(v2f a, v2f b, v8f c) {
  // D = A(16x4,f32) * B(4x16,f32) + C(16x16,f32)
  return __builtin_amdgcn_wmma_f32_16x16x4_f32(false, a, false, b, (short)0, c, false, false);
}

// ---------------------------------------------------------------------------
// Kernel 1: fused Q/K/V projection.
//   q[b,t,d] = (sum_c x[b,c,t]*Wq[c,d] + bq[d]) / sqrt(K)   (sel 0, [B,T,C])
//   k       from c with Wk                                  (sel 1, [B,T,C])
//   v       from c with Wv, stored TRANSPOSED [B,H,K,T]     (sel 2)
// One wave per 16x16 tile.
// ---------------------------------------------------------------------------
__global__ void __launch_bounds__(32)
qkv_proj_kernel(const float* __restrict__ x, const float* __restrict__ cin,
                const float* __restrict__ Wq, const float* __restrict__ bq,
                const float* __restrict__ Wk, const float* __restrict__ bk,
                const float* __restrict__ Wv, const float* __restrict__ bv,
                float* __restrict__ qo, float* __restrict__ ko, float* __restrict__ vt) {
  const int sel  = blockIdx.y;
  const int tile = blockIdx.x;
  const int ntc  = CCH / 16;               // 12
  const int ntt  = TSEQ / 16;              // 128
  const int dtile = tile % ntc;
  const int ttile = (tile / ntc) % ntt;
  const int b     = tile / (ntc * ntt);

  const int lane = threadIdx.x & 31;
  const int half = lane >> 4;
  const int m    = lane & 15;

  const float* A    = ((sel == 0) ? x : cin) + (size_t)b * CCH * TSEQ;  // [C,T]
  const float* W    = (sel == 0) ? Wq : (sel == 1) ? Wk : Wv;           // [C,C]
  const float* bias = (sel == 0) ? bq : (sel == 1) ? bk : bv;

  const int t0 = ttile * 16, d0 = dtile * 16;
  v8f acc = {};
  for (int k0 = 0; k0 < CCH; k0 += 4) {
    const int ka = k0 + 2 * half;
    v2f av, bv2;
    av.x  = A[(size_t)ka * TSEQ + t0 + m];        // A[t0+m][ka]   (x is [C,T])
    av.y  = A[(size_t)(ka + 1) * TSEQ + t0 + m];
    bv2.x = W[(size_t)ka * CCH + d0 + m];          // B[ka][d0+m]
    bv2.y = W[(size_t)(ka + 1) * CCH + d0 + m];
    acc = wmma_f32x4(av, bv2, acc);
  }

  const int d = d0 + m;
  if (sel == 2) {
    // V transposed: per lane the 8 outputs are contiguous in t -> 2x b128 store
    const int h2 = d / KH, kk = d % KH;
    const float bb = bias[d];
    float* vrow = vt + ((size_t)(b * NH + h2) * KH + kk) * TSEQ + t0 + 8 * half;
    v4f lo = {acc[0] + bb, acc[1] + bb, acc[2] + bb, acc[3] + bb};
    v4f hi = {acc[4] + bb, acc[5] + bb, acc[6] + bb, acc[7] + bb};
    *(v4f*)vrow       = lo;
    *(v4f*)(vrow + 4) = hi;
  } else {
    float* out = (sel == 0) ? qo : ko;
    const float scale = (sel == 0) ? QSCALE : 1.0f;
    const float bb = bias[d];
    for (int r = 0; r < 8; ++r) {
      const int t = t0 + r + 8 * half;
      out[((size_t)b * TSEQ + t) * CCH + d] = (acc[r] + bb) * scale;
    }
  }
}

// ---------------------------------------------------------------------------
// Kernel 2: rel-k logit table  rl[b,h,t,dd] = qs[b,h,t,:] . emb_rel_k[dd,:]
// (only |d|<=4 entries of the skew matrix are non-zero)
// ---------------------------------------------------------------------------
__global__ void rel_logits_kernel(const float* __restrict__ q,
                                  const float* __restrict__ erk,
                                  float* __restrict__ rl, int B) {
  const int idx = blockIdx.x * blockDim.x + threadIdx.x;   // (b*NH+h)*T + t
  if (idx >= B * NH * TSEQ) return;
  const int t = idx % TSEQ;
  const int h = (idx / TSEQ) % NH;
  const int b = idx / (TSEQ * NH);
  const v4f* q4 = (const v4f*)(q + ((size_t)b * TSEQ + t) * CCH + h * KH);
  for (int dd = 0; dd < 9; ++dd) {
    const v4f* e4 = (const v4f*)(erk + dd * KH);
    float s = 0.f;
    for (int kk = 0; kk < KH / 4; ++kk) {
      v4f a = q4[kk], e = e4[kk];
      s += a.x * e.x + a.y * e.y + a.z * e.z + a.w * e.w;
    }
    rl[(size_t)idx * 9 + dd] = s;
  }
}

// ---------------------------------------------------------------------------
// Kernel 3: banded attention. One wave per (b, h, 16-row tile).
// Columns [i0-256, i0+271]; out-of-range -> -1e30 (p==0 exactly),
// masked/out-of-block -> -1e4 (underflows in softmax, matching reference).
// ---------------------------------------------------------------------------
__global__ void __launch_bounds__(32)
banded_attn_kernel(const float* __restrict__ q, const float* __restrict__ kbuf,
                   const float* __restrict__ vt, const float* __restrict__ rl,
                   const int* __restrict__ mask, const float* __restrict__ relv,
                   float* __restrict__ ao) {
  __shared__ float S[16][WCOL];
  __shared__ float Qs[16][KH];
  __shared__ float rowInv[16];
  __shared__ float pm[2][16];
  __shared__ float ps[2][16];

  const int blk = blockIdx.x;                    // (b*NH+h)*(T/16) + it
  const int it  = blk % (TSEQ / 16);
  const int h   = (blk / (TSEQ / 16)) % NH;
  const int b   = blk / ((TSEQ / 16) * NH);
  const int i0  = it * 16;
  const int sbase = i0 - BAND;

  const int lane = threadIdx.x & 31;
  const int half = lane >> 4;
  const int m    = lane & 15;

  const float* Qg  = q    + (size_t)b * TSEQ * CCH + h * KH;           // [T,K]
  const float* Kg  = kbuf + (size_t)b * TSEQ * CCH + h * KH;           // [T,K]
  const float* Vt  = vt   + ((size_t)(b * NH + h) * KH) * TSEQ;        // [K,T]
  const float* rlb = rl   + ((size_t)(b * NH + h)) * TSEQ * 9;

  // stage 16x96 Q tile via float4
  for (int idx = lane; idx < 16 * (KH / 4); idx += 32) {
    const int rr = idx / (KH / 4), cc = (idx % (KH / 4)) * 4;
    *(v4f*)&Qs[rr][cc] = *(const v4f*)&Qg[(size_t)(i0 + rr) * CCH + cc];
  }
  __syncthreads();

  // ---- pass A: scores = qs . k^T  (+rel-k, -log1p|d|, mask) ----
  for (int j = 0; j < NJT; ++j) {
    const int s0 = sbase + j * 16;
    const int srow = s0 + m;
    const int scl = srow < 0 ? 0 : (srow >= TSEQ ? TSEQ - 1 : srow);
    const float* kr = Kg + (size_t)scl * CCH;
    v8f acc = {};
    for (int k0 = 0; k0 < KH; k0 += 4) {
      v2f av  = *(const v2f*)&Qs[m][k0 + 2 * half];   // ds_load_b64
      v2f bv2 = *(const v2f*)&kr[k0 + 2 * half];      // global_load_b64
      acc = wmma_f32x4(av, bv2, acc);
    }
    for (int r = 0; r < 8; ++r) {
      const int M = r + 8 * half;
      const int t = i0 + M;
      const int s = s0 + m;
      float sc;
      if (s < 0 || s >= TSEQ) {
        sc = -1e30f;                                   // exp -> exactly 0
      } else {
        sc = acc[r];
        const int d = s - t;
        if (d >= -4 && d <= 4) sc += rlb[(size_t)t * 9 + d + 4];
        sc -= log1pf(fabsf((float)d));
        const bool valid = (d >= -BAND && d <= BAND) &&
                           (mask[((size_t)b * TSEQ + t) * TSEQ + s] != 0);
        if (!valid) sc = -1e4f;
      }
      S[M][j * 16 + m] = sc;
    }
  }
  __syncthreads();

  // ---- softmax over the 528-wide band: lane (m,half) owns half of row m ----
  {
    const int c0 = half * (WCOL / 2);
    float mx = -3e38f;
    for (int i = 0; i < WCOL / 2; ++i) mx = fmaxf(mx, S[m][c0 + i]);
    pm[half][m] = mx;
    __syncthreads();
    const float rowmax = fmaxf(pm[0][m], pm[1][m]);
    float l = 0.f;
    for (int i = 0; i < WCOL / 2; ++i) {
      const float p = expf(S[m][c0 + i] - rowmax);
      S[m][c0 + i] = p;
      l += p;
    }
    ps[half][m] = l;
    __syncthreads();
    if (lane < 16) rowInv[m] = 1.0f / (ps[0][m] + ps[1][m]);
    __syncthreads();
  }

  // ---- pass B: O = P.V  (+ window-4 rel-v term), normalize, store ----
  for (int nb = 0; nb < KH / 16; ++nb) {
    const int n0 = nb * 16;
    const float* vrow = Vt + (size_t)(n0 + m) * TSEQ;   // V^T row for this lane's column
    v8f acc = {};
    for (int j = 0; j < NJT; ++j) {
      for (int kc = 0; kc < 4; ++kc) {
        const int lcol = j * 16 + kc * 4 + 2 * half;    // even
        int sr0 = sbase + lcol;                         // even; (sr0,sr0+1) uniformly valid
        sr0 = sr0 < 0 ? 0 : (sr0 > TSEQ - 2 ? TSEQ - 2 : sr0);
        v2f av  = *(const v2f*)&S[m][lcol];             // ds_load_b64 (p==0 if s OOR)
        v2f bv2 = *(const v2f*)&vrow[sr0];              // global_load_b64
        acc = wmma_f32x4(av, bv2, acc);
      }
    }
    for (int r = 0; r < 8; ++r) {
      const int M = r + 8 * half;
      const int t = i0 + M;
      const int kk = n0 + m;
      float rv = 0.f;
      for (int dd = 0; dd < 9; ++dd)
        rv += S[M][M + BAND + dd - 4] * relv[dd * KH + kk];
      ao[((size_t)b * TSEQ + t) * CCH + h * KH + kk] = (acc[r] + rv) * rowInv[M];
    }
  }
}

// ---------------------------------------------------------------------------
// Kernel 4: y[b,o,t] = sum_d ao[b,t,d]*Wo[d,o] + bo[o]  (store transposed)
// ---------------------------------------------------------------------------
__global__ void __launch_bounds__(32)
out_proj_kernel(const float* __restrict__ ao, const float* __restrict__ Wo,
                const float* __restrict__ bo, float* __restrict__ y) {
  const int tile = blockIdx.x;
  const int ntc = CCH / 16, ntt = TSEQ / 16;
  const int otile = tile % ntc;
  const int ttile = (tile / ntc) % ntt;
  const int b     = tile / (ntc * ntt);
  const int lane = threadIdx.x & 31;
  const int half = lane >> 4;
  const int m    = lane & 15;
  const int t0 = ttile * 16, o0 = otile * 16;
  const float* A = ao + (size_t)b * TSEQ * CCH;   // [T,C]

  v8f acc = {};
  for (int k0 = 0; k0 < CCH; k0 += 4) {
    const int ka = k0 + 2 * half;
    v2f av  = *(const v2f*)&A[(size_t)(t0 + m) * CCH + ka];  // global_load_b64
    v2f bv2;
    bv2.x = Wo[(size_t)ka * CCH + o0 + m];
    bv2.y = Wo[(size_t)(ka + 1) * CCH + o0 + m];
    acc = wmma_f32x4(av, bv2, acc);
  }
  const int o = o0 + m;
  const float bb = bo[o];
  for (int r = 0; r < 8; ++r) {
    const int t = t0 + r + 8 * half;
    y[((size_t)b * CCH + o) * TSEQ + t] = acc[r] + bb;
  }
}

// ---------------------------------------------------------------------------
extern "C" void kernel_launch(void* const* d_in, const int* in_sizes, int n_in,
                              void* d_out, int out_size, void* d_ws, size_t ws_size,
                              hipStream_t stream) {
  const float* x   = (const float*)d_in[0];
  const float* c   = (const float*)d_in[1];
  const float* Wq  = (const float*)d_in[2];
  const float* bq  = (const float*)d_in[3];
  const float* Wk  = (const float*)d_in[4];
  const float* bk  = (const float*)d_in[5];
  const float* Wv  = (const float*)d_in[6];
  const float* bv  = (const float*)d_in[7];
  const float* Wo  = (const float*)d_in[8];
  const float* bo  = (const float*)d_in[9];
  const float* erk = (const float*)d_in[10];
  const float* erv = (const float*)d_in[11];
  const int*  mask = (const int*)d_in[12];
  float* y = (float*)d_out;

  const int B = in_sizes[0] / (CCH * TSEQ);

  float* ws = (float*)d_ws;
  const size_t nmat = (size_t)B * TSEQ * CCH;
  float* q  = ws;
  float* kb = q  + nmat;
  float* vt = kb + nmat;          // V stored transposed [B,H,K,T]
  float* ao = vt + nmat;
  float* rl = ao + nmat;          // B*NH*TSEQ*9 floats

  const int ntiles = B * (TSEQ / 16) * (CCH / 16);
  dim3 g1(ntiles, 3);
  qkv_proj_kernel<<<g1, 32, 0, stream>>>(x, c, Wq, bq, Wk, bk, Wv, bv, q, kb, vt);

  const int tot = B * NH * TSEQ;
  rel_logits_kernel<<<(tot + 127) / 128, 128, 0, stream>>>(q, erk, rl, B);

  const int nattn = B * NH * (TSEQ / 16);
  banded_attn_kernel<<<nattn, 32, 0, stream>>>(q, kb, vt, rl, mask, erv, ao);

  out_proj_kernel<<<ntiles, 32, 0, stream>>>(ao, Wo, bo, y);
}